// TransNet_17952963298134
// MI455X (gfx1250) — compile-verified
//
#include <hip/hip_runtime.h>
#include <math.h>

typedef __attribute__((ext_vector_type(2))) float v2f;
typedef __attribute__((ext_vector_type(8))) float v8f;

#define BATCH  2048
#define ACTD   16
#define LATD   64
#define INPD   256
#define MIDD   256
#define GATES  1024
#define PARAMS 66048

__device__ __forceinline__ float sigmoidf_(float x) {
    return 1.0f / (1.0f + __expf(-x));
}

// ---------------------------------------------------------------------------
// Front-end: cond path + obs add + dense_in, all EvoNormS0 with group size 8.
// One block (256 threads) per batch row.
// ---------------------------------------------------------------------------
__global__ void frontend_kernel(const float* __restrict__ actions,
                                const float* __restrict__ obs,
                                const float* __restrict__ W_cond_in,
                                const float* __restrict__ v_ci,
                                const float* __restrict__ g_ci,
                                const float* __restrict__ b_ci,
                                const float* __restrict__ W_cond_lat,
                                const float* __restrict__ b_cl,
                                const float* __restrict__ W_in,
                                const float* __restrict__ v_in,
                                const float* __restrict__ g_in,
                                const float* __restrict__ b_in,
                                float* __restrict__ x1) {
    __shared__ float t_raw[LATD];
    __shared__ float t_nrm[LATD];
    __shared__ float x0[LATD];
    __shared__ float u_raw[INPD];
    const int row = blockIdx.x;
    const int tid = threadIdx.x;

    if (tid < LATD) {
        float acc = 0.0f;
        #pragma unroll
        for (int k = 0; k < ACTD; ++k)
            acc = fmaf(actions[row * ACTD + k], W_cond_in[k * LATD + tid], acc);
        t_raw[tid] = acc;
    }
    __syncthreads();
    if (tid < LATD) {
        const int g0 = tid & ~7;
        float m = 0.0f, m2 = 0.0f;
        #pragma unroll
        for (int j = 0; j < 8; ++j) { float v = t_raw[g0 + j]; m += v; m2 += v * v; }
        m *= 0.125f; m2 *= 0.125f;
        const float var = m2 - m * m;
        const float x = t_raw[tid];
        t_nrm[tid] = x * sigmoidf_(v_ci[tid] * x) * rsqrtf(var + 1e-5f) * g_ci[tid] + b_ci[tid];
    }
    __syncthreads();
    if (tid < LATD) {
        float acc = b_cl[tid] + obs[row * LATD + tid];  // obs [8,8] flattened = 64
        #pragma unroll 8
        for (int k = 0; k < LATD; ++k)
            acc = fmaf(t_nrm[k], W_cond_lat[k * LATD + tid], acc);
        x0[tid] = acc;
    }
    __syncthreads();
    {
        float acc = 0.0f;
        #pragma unroll 8
        for (int k = 0; k < LATD; ++k)
            acc = fmaf(x0[k], W_in[k * INPD + tid], acc);
        u_raw[tid] = acc;
    }
    __syncthreads();
    {
        const int g0 = tid & ~7;
        float m = 0.0f, m2 = 0.0f;
        #pragma unroll
        for (int j = 0; j < 8; ++j) { float v = u_raw[g0 + j]; m += v; m2 += v * v; }
        m *= 0.125f; m2 *= 0.125f;
        const float var = m2 - m * m;
        const float x = u_raw[tid];
        x1[(size_t)row * INPD + tid] =
            x * sigmoidf_(v_in[tid] * x) * rsqrtf(var + 1e-5f) * g_in[tid] + b_in[tid];
    }
}

// ---------------------------------------------------------------------------
// fp32 WMMA GEMM: C[M,N] = A[M,K] @ B[K,N] (+ bias), row-major, K == 256.
// Block = 256 threads = 8 waves. Each block computes a 32x512 output patch:
//   - 2 M-tiles (32 rows), A staged in LDS as one contiguous float4 memcpy,
//     shared by all 8 waves.
//   - each wave owns 4 N-tiles (64 cols) -> 8 v8f accumulators per wave.
// Per k-step of 4: 2 LDS A-frag loads + 8 B loads feed 8 WMMAs.
// Requirements: M % 32 == 0, N % 512 == 0 (2048, 1024, 66048 all satisfy).
// Fragment layout (ISA 7.12.2):
//   A 16x4 f32: lane&15 = M row, lane>>4 selects K pair; 2 VGPRs (K, K+1).
//   B 4x16 f32: lane&15 = N col, lane>>4 selects K pair; 2 VGPRs.
//   C/D 16x16: VGPR v -> M = v + 8*(lane>>4), N = lane&15.
// ---------------------------------------------------------------------------
__global__ void gemm_wmma_kernel(const float* __restrict__ A,
                                 const float* __restrict__ Bm,
                                 const float* __restrict__ bias,
                                 float* __restrict__ C,
                                 int M, int N, int K) {
    __shared__ float As[32 * 256];  // 32 KB: two 16-row A tiles
    const int tid    = threadIdx.x;
    const int m_base = blockIdx.y * 32;

    // A tile = 32 consecutive full rows -> contiguous block; vectorized copy.
    {
        const float4* A4  = (const float4*)(A + (size_t)m_base * K);
        float4*       As4 = (float4*)As;
        const int     n4  = (32 * K) >> 2;   // 2048 float4s
        for (int idx = tid; idx < n4; idx += 256)
            As4[idx] = A4[idx];
    }
    __syncthreads();

    const int lane   = tid & 31;
    const int wv     = tid >> 5;
    const int n_base = blockIdx.x * 512 + wv * 64;  // this wave's 4 N-tiles
    const int ln     = lane & 15;   // A: M row, B: N col
    const int khalf  = lane >> 4;   // selects K pair {0,1} vs {2,3}

    v8f acc[2][4] = {};

    const float* Bp = Bm + (size_t)(2 * khalf) * N + n_base + ln;

    for (int k = 0; k < K; k += 4) {
        // A fragments for the two M tiles (LDS, 2-dword loads)
        v2f a0, a1;
        a0.x = As[ln * K + k + 2 * khalf];
        a0.y = As[ln * K + k + 2 * khalf + 1];
        a1.x = As[(16 + ln) * K + k + 2 * khalf];
        a1.y = As[(16 + ln) * K + k + 2 * khalf + 1];

        const float* Bk = Bp + (size_t)k * N;
        // prefetch B rows two k-steps ahead (global_prefetch_b8)
        __builtin_prefetch(Bk + (size_t)8 * N, 0, 0);

        #pragma unroll
        for (int j = 0; j < 4; ++j) {
            v2f b;
            b.x = Bk[j * 16];
            b.y = Bk[(size_t)N + j * 16];
            acc[0][j] = __builtin_amdgcn_wmma_f32_16x16x4_f32(
                false, a0, false, b, (short)0, acc[0][j], false, false);
            acc[1][j] = __builtin_amdgcn_wmma_f32_16x16x4_f32(
                false, a1, false, b, (short)0, acc[1][j], false, false);
        }
    }

    #pragma unroll
    for (int mt = 0; mt < 2; ++mt) {
        #pragma unroll
        for (int j = 0; j < 4; ++j) {
            const int  ncol = n_base + j * 16 + ln;
            const float bval = bias ? bias[ncol] : 0.0f;
            #pragma unroll
            for (int v = 0; v < 8; ++v) {
                const int r = m_base + mt * 16 + v + khalf * 8;
                C[(size_t)r * N + ncol] = acc[mt][j][v] + bval;
            }
        }
    }
}

// ---------------------------------------------------------------------------
// Sequential LSTM scan: 2048 steps; Z = x1@Wk precomputed. One workgroup of
// 1024 threads; h/c live in LDS; Wr (1 MB) streamed from L2 each step.
// Thread j computes gate column j: z[j] = Z[t][j] + sum_k h[k]*Wr[k][j].
// ---------------------------------------------------------------------------
__global__ void lstm_kernel(const float* __restrict__ Z,
                            const float* __restrict__ Wr,
                            const float* __restrict__ v_l,
                            const float* __restrict__ g_l,
                            const float* __restrict__ b_l,
                            float* __restrict__ H) {
    __shared__ float h_lds[MIDD];
    __shared__ float c_lds[MIDD];
    __shared__ float zbuf[GATES];
    const int tid = threadIdx.x;

    if (tid < MIDD) { h_lds[tid] = 0.0f; c_lds[tid] = 0.0f; }
    __syncthreads();

    for (int t = 0; t < BATCH; ++t) {
        float acc = Z[(size_t)t * GATES + tid];
        #pragma unroll 8
        for (int k = 0; k < MIDD; ++k)
            acc = fmaf(h_lds[k], Wr[k * GATES + tid], acc);  // h: LDS broadcast
        zbuf[tid] = acc;
        __syncthreads();

        float o_g = 0.0f;
        if (tid < MIDD) {
            const float i_g = zbuf[tid];
            const float f_g = zbuf[MIDD + tid];
            const float cc  = zbuf[2 * MIDD + tid];
            o_g             = zbuf[3 * MIDD + tid];
            // EvoNormS0 of cc gate (group size 8)
            const int g0 = 2 * MIDD + (tid & ~7);
            float m = 0.0f, m2 = 0.0f;
            #pragma unroll
            for (int j = 0; j < 8; ++j) { float v = zbuf[g0 + j]; m += v; m2 += v * v; }
            m *= 0.125f; m2 *= 0.125f;
            const float var = m2 - m * m;
            const float cn =
                cc * sigmoidf_(v_l[tid] * cc) * rsqrtf(var + 1e-5f) * g_l[tid] + b_l[tid];
            c_lds[tid] = sigmoidf_(f_g) * c_lds[tid] + sigmoidf_(i_g) * cn;
        }
        __syncthreads();
        if (tid < MIDD) {
            // EvoNormS0 of c_new (group size 8)
            const int g0 = tid & ~7;
            float m = 0.0f, m2 = 0.0f;
            #pragma unroll
            for (int j = 0; j < 8; ++j) { float v = c_lds[g0 + j]; m += v; m2 += v * v; }
            m *= 0.125f; m2 *= 0.125f;
            const float var = m2 - m * m;
            const float c_new = c_lds[tid];
            const float cn =
                c_new * sigmoidf_(v_l[tid] * c_new) * rsqrtf(var + 1e-5f) * g_l[tid] + b_l[tid];
            const float h_new = sigmoidf_(o_g) * cn;
            h_lds[tid] = h_new;
            H[(size_t)t * MIDD + tid] = h_new;
        }
        __syncthreads();
    }
}

// ---------------------------------------------------------------------------
// dense_0 + EvoNormS0: one block (256 threads) per row.
// ---------------------------------------------------------------------------
__global__ void dense0_kernel(const float* __restrict__ H,
                              const float* __restrict__ W_d0,
                              const float* __restrict__ v_d0,
                              const float* __restrict__ g_d0,
                              const float* __restrict__ b_d0,
                              float* __restrict__ x2) {
    __shared__ float hrow[MIDD];
    __shared__ float u[MIDD];
    const int row = blockIdx.x;
    const int tid = threadIdx.x;
    hrow[tid] = H[(size_t)row * MIDD + tid];
    __syncthreads();
    float acc = 0.0f;
    #pragma unroll 8
    for (int k = 0; k < MIDD; ++k)
        acc = fmaf(hrow[k], W_d0[k * MIDD + tid], acc);
    u[tid] = acc;
    __syncthreads();
    const int g0 = tid & ~7;
    float m = 0.0f, m2 = 0.0f;
    #pragma unroll
    for (int j = 0; j < 8; ++j) { float v = u[g0 + j]; m += v; m2 += v * v; }
    m *= 0.125f; m2 *= 0.125f;
    const float var = m2 - m * m;
    const float x = u[tid];
    x2[(size_t)row * MIDD + tid] =
        x * sigmoidf_(v_d0[tid] * x) * rsqrtf(var + 1e-5f) * g_d0[tid] + b_d0[tid];
}

// ---------------------------------------------------------------------------
extern "C" void kernel_launch(void* const* d_in, const int* in_sizes, int n_in,
                              void* d_out, int out_size, void* d_ws, size_t ws_size,
                              hipStream_t stream) {
    const float* actions    = (const float*)d_in[0];
    const float* obs        = (const float*)d_in[1];
    const float* W_cond_in  = (const float*)d_in[2];
    const float* v_ci       = (const float*)d_in[3];
    const float* g_ci       = (const float*)d_in[4];
    const float* b_ci       = (const float*)d_in[5];
    const float* W_cond_lat = (const float*)d_in[6];
    const float* b_cl       = (const float*)d_in[7];
    const float* W_in       = (const float*)d_in[8];
    const float* v_in       = (const float*)d_in[9];
    const float* g_in       = (const float*)d_in[10];
    const float* b_in       = (const float*)d_in[11];
    const float* Wk         = (const float*)d_in[12];
    const float* Wr         = (const float*)d_in[13];
    const float* v_l        = (const float*)d_in[14];
    const float* g_l        = (const float*)d_in[15];
    const float* b_l        = (const float*)d_in[16];
    const float* W_d0       = (const float*)d_in[17];
    const float* v_d0       = (const float*)d_in[18];
    const float* g_d0       = (const float*)d_in[19];
    const float* b_d0       = (const float*)d_in[20];
    const float* W_out      = (const float*)d_in[21];
    const float* b_out      = (const float*)d_in[22];
    float* out = (float*)d_out;

    // Workspace layout (floats): x1[2048*256] | Z[2048*1024] | H[2048*256] | x2[2048*256]
    float* ws = (float*)d_ws;
    float* x1 = ws;
    float* Z  = x1 + (size_t)BATCH * INPD;
    float* H  = Z  + (size_t)BATCH * GATES;
    float* x2 = H  + (size_t)BATCH * MIDD;

    frontend_kernel<<<BATCH, 256, 0, stream>>>(actions, obs, W_cond_in, v_ci, g_ci, b_ci,
                                               W_cond_lat, b_cl, W_in, v_in, g_in, b_in, x1);

    // Z = x1 @ Wk  : [2048,256] x [256,1024]  (hoisted out of the scan)
    dim3 gz(GATES / 512, BATCH / 32);      // (2, 64)
    gemm_wmma_kernel<<<gz, 256, 0, stream>>>(x1, Wk, nullptr, Z, BATCH, GATES, INPD);

    lstm_kernel<<<1, 1024, 0, stream>>>(Z, Wr, v_l, g_l, b_l, H);

    dense0_kernel<<<BATCH, 256, 0, stream>>>(H, W_d0, v_d0, g_d0, b_d0, x2);

    // out = x2 @ W_out + b_out : [2048,256] x [256,66048]
    dim3 go(PARAMS / 512, BATCH / 32);     // (129, 64)
    gemm_wmma_kernel<<<go, 256, 0, stream>>>(x2, W_out, b_out, out, BATCH, PARAMS, INPD);
}